// MultiHeadNet_764504178772
// MI455X (gfx1250) — compile-verified
//
#include <hip/hip_runtime.h>

typedef __attribute__((ext_vector_type(2))) float v2f;
typedef __attribute__((ext_vector_type(8))) float v8f;

#define WMMA_F32(a, b, c) \
  __builtin_amdgcn_wmma_f32_16x16x4_f32(false, (a), false, (b), (short)0, (c), false, false)

constexpr int BATCH  = 8192;
constexpr int NBITS  = 3;
constexpr int DFEAT  = 512;
constexpr int DBASE  = 1024;
constexpr int DHEAD  = 1024;
constexpr int DOUT   = 512;
constexpr int NHEADS = 8;
constexpr int XC     = NBITS + DFEAT;   // 515 columns in x

// ---------------------------------------------------------------- CDNA5 async helpers
__device__ __forceinline__ unsigned lds_off(const void* p) {
  // Generic pointers into LDS carry the wave-relative LDS byte offset in the low 32 bits.
  return (unsigned)(unsigned long long)p;
}
__device__ __forceinline__ void async_ld_b128(unsigned ldsDst, const void* gsrc) {
  asm volatile("global_load_async_to_lds_b128 %0, %1, off"
               :: "v"(ldsDst), "v"(gsrc) : "memory");
}
__device__ __forceinline__ void async_ld_b32(unsigned ldsDst, const void* gsrc) {
  asm volatile("global_load_async_to_lds_b32 %0, %1, off"
               :: "v"(ldsDst), "v"(gsrc) : "memory");
}
template <int N>
__device__ __forceinline__ void wait_async() {
  asm volatile("s_wait_asynccnt %0" :: "n"(N) : "memory");
}

// ---------------------------------------------------------------- routing ---
__global__ void init_counts_kernel(int* __restrict__ counts) {
  if (threadIdx.x < NHEADS) counts[threadIdx.x] = 0;
}

__global__ void router_kernel(const float* __restrict__ x,
                              int* __restrict__ counts,
                              int* __restrict__ buckets) {
  int r = blockIdx.x * blockDim.x + threadIdx.x;
  if (r >= BATCH) return;
  const float* xr = x + (size_t)r * XC;
  int h = (xr[0] > 0.5f ? 1 : 0) | (xr[1] > 0.5f ? 2 : 0) | (xr[2] > 0.5f ? 4 : 0);
  int slot = atomicAdd(&counts[h], 1);
  buckets[h * BATCH + slot] = r;
}

// -------------------------------------------------- base = relu(x@Wb + bb) ---
// Block = 512 threads (16 waves). Block tile: 64 rows x 256 cols.
// Wave tile: 64x16 (4 row-groups x 1 col-tile): B loaded once per k-step,
// reused by 4 WMMAs. A (x rows) staged into LDS with async b32 copies,
// double-buffered in 16-wide K chunks.
constexpr int BG_KC = 16;              // K chunk
constexpr int BG_SX = BG_KC + 4;       // LDS row stride (floats): conflict-free
__global__ __launch_bounds__(512) void base_gemm_kernel(
    const float* __restrict__ x, const float* __restrict__ Wb,
    const float* __restrict__ bb, float* __restrict__ base) {
  __shared__ float sX[2][64 * BG_SX];  // 10 KB

  const int tid    = threadIdx.x;
  const int lane   = tid & 31;
  const int wave   = tid >> 5;
  const int lanelo = lane & 15;
  const int khalf  = lane >> 4;
  const int rowTile = blockIdx.y * 64;
  const int col     = blockIdx.x * 256 + wave * 16 + lanelo;

  // Staging: 64 rows x 16 k = 1024 floats / 512 threads = 2 floats each.
  const int sr  = tid >> 3;                       // 0..63
  const int skq = (tid & 7) * 2;                  // 0..14
  const float* sgp = x + (size_t)(rowTile + sr) * XC + NBITS + skq;
  const unsigned sd0 = lds_off(&sX[0][sr * BG_SX + skq]);
  const unsigned bufB = (unsigned)(64 * BG_SX * sizeof(float));

  v8f acc[4] = {};
  // prefetch chunk 0
  async_ld_b32(sd0, sgp);
  async_ld_b32(sd0 + 4, sgp + 1);
  const int nchunks = DFEAT / BG_KC;              // 32
  for (int c = 0; c < nchunks; ++c) {
    if (c + 1 < nchunks) {
      const unsigned d = sd0 + ((c + 1) & 1) * bufB;
      const float* g = sgp + (c + 1) * BG_KC;
      async_ld_b32(d, g);
      async_ld_b32(d + 4, g + 1);
      wait_async<2>();
    } else {
      wait_async<0>();
    }
    __syncthreads();
    const float* cur = sX[c & 1];
    #pragma unroll
    for (int kk = 0; kk < BG_KC; kk += 4) {
      const int kloc = kk + 2 * khalf;
      const int kg   = c * BG_KC + kloc;
      v2f b;
      b.x = Wb[(size_t)kg * DBASE + col];
      b.y = Wb[(size_t)(kg + 1) * DBASE + col];
      #pragma unroll
      for (int rg = 0; rg < 4; ++rg) {
        v2f a;
        a.x = cur[(rg * 16 + lanelo) * BG_SX + kloc];
        a.y = cur[(rg * 16 + lanelo) * BG_SX + kloc + 1];
        acc[rg] = WMMA_F32(a, b, acc[rg]);
      }
    }
    __syncthreads();
  }
  const float bias = bb[col];
  #pragma unroll
  for (int rg = 0; rg < 4; ++rg)
    #pragma unroll
    for (int v = 0; v < 8; ++v) {
      const int row = rowTile + rg * 16 + 8 * khalf + v;
      base[(size_t)row * DBASE + col] = fmaxf(acc[rg][v] + bias, 0.0f);
    }
}

// ------------------------------------- fused per-head MLP on routed rows -----
// Block = 512 threads (16 waves), one block per (32-row bucket tile, head).
// Phase 1: h[32x1024] = relu(base_rows @ Wh1[head] + bh1) -> LDS (k-major, pad 33).
//          Wave tile 32x64 (2 rg x 4 ct). A panel async-staged (b128, dbl-buffered).
// Phase 2: out[32x512] = h @ Wh2[head] + bh2, wave tile 32x32, scatter stores.
constexpr int MT      = 32;            // rows per block tile
constexpr int HM_KC   = 64;            // k-chunk for A staging
constexpr int SA_STR  = HM_KC + 4;     // 68 floats = 272B: 16B-aligned, conflict-free
constexpr int HB_STR  = MT + 1;        // 33: conflict-free writes & reads
__global__ __launch_bounds__(512) void head_mlp_kernel(
    const float* __restrict__ base,
    const float* __restrict__ Wh1, const float* __restrict__ bh1,
    const float* __restrict__ Wh2, const float* __restrict__ bh2,
    const int* __restrict__ counts, const int* __restrict__ buckets,
    float* __restrict__ out) {
  extern __shared__ float smem[];
  float* hbuf = smem;                          // DHEAD * 33 floats (132 KB)
  float* sA   = smem + DHEAD * HB_STR;         // 2 * 32 * 68 floats (17 KB)

  const int head = blockIdx.y;
  const int cnt  = counts[head];
  const int t0   = blockIdx.x * MT;
  if (t0 >= cnt) return;                       // uniform per block
  const int rem = min(MT, cnt - t0);

  const int tid    = threadIdx.x;
  const int lane   = tid & 31;
  const int wave   = tid >> 5;
  const int lanelo = lane & 15;
  const int khalf  = lane >> 4;
  const int* myBucket = buckets + head * BATCH + t0;

  // Staging: 32 rows x 64 k x 4B = 8 KB -> 512 threads x one b128 each.
  const int sr  = tid >> 4;                    // 0..31 (row in tile)
  const int skq = (tid & 15) * 4;              // 0..60 (k within chunk)
  const int srow = myBucket[sr < rem ? sr : rem - 1];   // clamp padded rows
  const float* sgp = base + (size_t)srow * DBASE + skq;
  const unsigned sd0  = lds_off(sA) + (unsigned)((sr * SA_STR + skq) * sizeof(float));
  const unsigned bufB = (unsigned)(MT * SA_STR * sizeof(float));

  // ---- Phase 1: wave covers h cols [wave*64, wave*64+64) ----
  const float* W1 = Wh1 + (size_t)head * DBASE * DHEAD;
  const int colBase = wave * 64;
  {
    v8f acc[2][4] = {};
    async_ld_b128(sd0, sgp);                   // prefetch chunk 0
    const int nchunks = DBASE / HM_KC;         // 16
    for (int c = 0; c < nchunks; ++c) {
      if (c + 1 < nchunks) {
        async_ld_b128(sd0 + ((c + 1) & 1) * bufB, sgp + (c + 1) * HM_KC);
        wait_async<1>();
      } else {
        wait_async<0>();
      }
      __syncthreads();
      const float* cur = sA + (c & 1) * (MT * SA_STR);
      #pragma unroll 4
      for (int kk = 0; kk < HM_KC; kk += 4) {
        const int kloc = kk + 2 * khalf;
        const int kg   = c * HM_KC + kloc;
        v2f a0, a1;
        a0.x = cur[lanelo * SA_STR + kloc];
        a0.y = cur[lanelo * SA_STR + kloc + 1];
        a1.x = cur[(lanelo + 16) * SA_STR + kloc];
        a1.y = cur[(lanelo + 16) * SA_STR + kloc + 1];
        #pragma unroll
        for (int s = 0; s < 4; ++s) {
          const int col = colBase + s * 16 + lanelo;
          v2f b;
          b.x = W1[(size_t)kg * DHEAD + col];
          b.y = W1[(size_t)(kg + 1) * DHEAD + col];
          acc[0][s] = WMMA_F32(a0, b, acc[0][s]);
          acc[1][s] = WMMA_F32(a1, b, acc[1][s]);
        }
      }
      __syncthreads();
    }
    #pragma unroll
    for (int s = 0; s < 4; ++s) {
      const int col = colBase + s * 16 + lanelo;
      const float bias = bh1[head * DHEAD + col];
      #pragma unroll
      for (int rg = 0; rg < 2; ++rg)
        #pragma unroll
        for (int v = 0; v < 8; ++v) {
          const int m = rg * 16 + 8 * khalf + v;
          hbuf[col * HB_STR + m] = fmaxf(acc[rg][s][v] + bias, 0.0f);
        }
    }
  }
  __syncthreads();

  // ---- Phase 2: wave covers out cols [wave*32, wave*32+32) ----
  {
    const float* W2 = Wh2 + (size_t)head * DHEAD * DOUT;
    const int colBase2 = wave * 32;
    v8f acc[2][2] = {};
    #pragma unroll 4
    for (int k0 = 0; k0 < DHEAD; k0 += 4) {
      const int ka = k0 + 2 * khalf;
      v2f a0, a1;
      a0.x = hbuf[ka * HB_STR + lanelo];
      a0.y = hbuf[(ka + 1) * HB_STR + lanelo];
      a1.x = hbuf[ka * HB_STR + lanelo + 16];
      a1.y = hbuf[(ka + 1) * HB_STR + lanelo + 16];
      #pragma unroll
      for (int s = 0; s < 2; ++s) {
        const int col = colBase2 + s * 16 + lanelo;
        v2f b;
        b.x = W2[(size_t)ka * DOUT + col];
        b.y = W2[(size_t)(ka + 1) * DOUT + col];
        acc[0][s] = WMMA_F32(a0, b, acc[0][s]);
        acc[1][s] = WMMA_F32(a1, b, acc[1][s]);
      }
    }
    #pragma unroll
    for (int s = 0; s < 2; ++s) {
      const int col = colBase2 + s * 16 + lanelo;
      const float bias = bh2[head * DOUT + col];
      #pragma unroll
      for (int rg = 0; rg < 2; ++rg)
        #pragma unroll
        for (int v = 0; v < 8; ++v) {
          const int m = rg * 16 + 8 * khalf + v;
          if (m < rem) {                       // tail mask: divergence only on stores
            const int row = myBucket[m];
            out[(size_t)row * DOUT + col] = acc[rg][s][v] + bias;
          }
        }
    }
  }
}

// ---------------------------------------------------------------- launch -----
extern "C" void kernel_launch(void* const* d_in, const int* in_sizes, int n_in,
                              void* d_out, int out_size, void* d_ws, size_t ws_size,
                              hipStream_t stream) {
  (void)in_sizes; (void)n_in; (void)out_size; (void)ws_size;
  const float* x   = (const float*)d_in[0];
  const float* Wb  = (const float*)d_in[1];
  const float* bb  = (const float*)d_in[2];
  const float* Wh1 = (const float*)d_in[3];
  const float* bh1 = (const float*)d_in[4];
  const float* Wh2 = (const float*)d_in[5];
  const float* bh2 = (const float*)d_in[6];
  float* out = (float*)d_out;

  // Workspace layout: base[8192*1024 f32] | counts[16 int] | buckets[8*8192 int]
  float* base    = (float*)d_ws;
  int*   counts  = (int*)((char*)d_ws + (size_t)BATCH * DBASE * sizeof(float));
  int*   buckets = counts + 16;

  init_counts_kernel<<<1, 32, 0, stream>>>(counts);
  router_kernel<<<BATCH / 256, 256, 0, stream>>>(x, counts, buckets);
  base_gemm_kernel<<<dim3(DBASE / 256, BATCH / 64), 512, 0, stream>>>(x, Wb, bb, base);

  const size_t smemBytes = (size_t)(DHEAD * HB_STR + 2 * MT * SA_STR) * sizeof(float);
  head_mlp_kernel<<<dim3(BATCH / MT, NHEADS), 512, smemBytes, stream>>>(
      base, Wh1, bh1, Wh2, bh2, counts, buckets, out);
}